// LaplacianBuilder_43662637531924
// MI455X (gfx1250) — compile-verified
//
#include <hip/hip_runtime.h>

typedef __attribute__((ext_vector_type(2))) float v2f;
typedef __attribute__((ext_vector_type(4))) float v4f;
typedef __attribute__((ext_vector_type(8))) float v8f;

// ---------------------------------------------------------------------------
// K1: stream-zero the 576MB output with non-temporal B128 stores.
// ---------------------------------------------------------------------------
__global__ void zero_out_kernel(float* __restrict__ out, long long n4) {
  long long i = (long long)blockIdx.x * blockDim.x + threadIdx.x;
  long long stride = (long long)gridDim.x * blockDim.x;
  v4f* p = (v4f*)out;
  v4f z = {0.f, 0.f, 0.f, 0.f};
  for (; i < n4; i += stride)
    __builtin_nontemporal_store(z, p + i);
}

__global__ void zero_ws_kernel(float* __restrict__ p, int n) {
  int i = blockIdx.x * blockDim.x + threadIdx.x;
  if (i < n) p[i] = 0.f;
}

// ---------------------------------------------------------------------------
// K2: per-edge Gram M^T M via V_WMMA_F32_16X16X4_F32, 4 edges per wave.
//   A (16x4) = [M0^T; M1^T; M2^T; M3^T], B (4x16) = [M0 M1 M2 M3]
//   -> diagonal 4x4 blocks of the 16x16 D matrix are M_e^T M_e.
// Per ISA layouts the per-lane loads for A and B coincide, so one fetch
// feeds both operands. Results are atomically segment-summed into diag[row].
// ---------------------------------------------------------------------------
__global__ void gram_diag_kernel(const float* __restrict__ maps,
                                 const int* __restrict__ eidx,
                                 float* __restrict__ diag,
                                 int twoE) {
  int tid = blockIdx.x * blockDim.x + threadIdx.x;
  int wave = tid >> 5;
  int lane = tid & 31;
  int ebase = wave * 4;
  if (ebase >= twoE) return;          // wave-uniform: EXEC all-1 for WMMA

  int n16 = lane & 15;
  int ep = n16 >> 2;                  // local edge 0..3 within wave
  int i  = n16 & 3;                   // intra-block index
  int kb = (lane < 16) ? 0 : 2;       // K pair held by this half-wave

  int esrc = ebase + ep;
  if (esrc > twoE - 1) esrc = twoE - 1;   // clamped load; result discarded
  const float* m = maps + (size_t)esrc * 16;

  // A[16x4] row r=4*ep+i, cols kb,kb+1 : A[r][k] = (M^T)[i][k] = M[k][i]
  v2f a;
  a.x = m[kb * 4 + i];
  a.y = m[(kb + 1) * 4 + i];

  v8f c = {};
  c = __builtin_amdgcn_wmma_f32_16x16x4_f32(
      /*neg_a=*/false, a, /*neg_b=*/false, a,
      /*c_mod=*/(short)0, c, /*reuse_a=*/false, /*reuse_b=*/false);

  // Constant-index extracts first (avoids dynamic-extract cndmask cascades),
  // then plain scalar selects on one VCC.
  float c0 = c[0], c1 = c[1], c2 = c[2], c3 = c[3];
  float c4 = c[4], c5 = c[5], c6 = c[6], c7 = c[7];

  // Diagonal block (ep,ep) of D lives in lanes 0-7 (ep=0,1) and 24-31 (ep=2,3),
  // at VGPRs vbase..vbase+3 with vbase = 4*(ep&1). Column index j = i.
  bool useful = (lane < 8) || (lane >= 24);
  if (useful && (ebase + ep) < twoE) {
    bool hi = (ep & 1) != 0;
    float g0 = hi ? c4 : c0;
    float g1 = hi ? c5 : c1;
    float g2 = hi ? c6 : c2;
    float g3 = hi ? c7 : c3;
    int j = i;
    int row = eidx[ebase + ep];
    float* dst = diag + (size_t)row * 16;
    atomicAdd(dst + 0 * 4 + j, g0);
    atomicAdd(dst + 1 * 4 + j, g1);
    atomicAdd(dst + 2 * 4 + j, g2);
    atomicAdd(dst + 3 * 4 + j, g3);
  }
}

// ---------------------------------------------------------------------------
// 4x4 register matmuls (fully unrolled -> pure VALU FMA).
// ---------------------------------------------------------------------------
__device__ __forceinline__ void mm4(const float* __restrict__ A,
                                    const float* __restrict__ B,
                                    float* __restrict__ C) {
#pragma unroll
  for (int i = 0; i < 4; ++i)
#pragma unroll
    for (int j = 0; j < 4; ++j) {
      float s = 0.f;
#pragma unroll
      for (int k = 0; k < 4; ++k) s = fmaf(A[i * 4 + k], B[k * 4 + j], s);
      C[i * 4 + j] = s;
    }
}

__device__ __forceinline__ void mm4_tn(const float* __restrict__ A,
                                       const float* __restrict__ B,
                                       float* __restrict__ C) {
  // C = A^T * B
#pragma unroll
  for (int i = 0; i < 4; ++i)
#pragma unroll
    for (int j = 0; j < 4; ++j) {
      float s = 0.f;
#pragma unroll
      for (int k = 0; k < 4; ++k) s = fmaf(A[k * 4 + i], B[k * 4 + j], s);
      C[i * 4 + j] = s;
    }
}

// ---------------------------------------------------------------------------
// K3: per-node (to_inv = diag + I)^{-1/2} via trace-normalized coupled
// Newton-Schulz, then write clipped diagonal block S*G*S into the output.
// ---------------------------------------------------------------------------
__global__ void invsqrt_diag_kernel(const float* __restrict__ diag,
                                    float* __restrict__ invs,
                                    float* __restrict__ out,
                                    int n, int N) {
  int node = blockIdx.x * blockDim.x + threadIdx.x;
  if (node >= n) return;

  float G[16], Y[16], Z[16], W[16], T[16];
  const float* g = diag + (size_t)node * 16;
#pragma unroll
  for (int k = 0; k < 16; ++k) G[k] = g[k];

  // A = G + I, scaled by 1/trace so eigenvalues land in (0,1]
  float tr = (G[0] + 1.f) + (G[5] + 1.f) + (G[10] + 1.f) + (G[15] + 1.f);
  float sinv = 1.f / tr;
#pragma unroll
  for (int k = 0; k < 16; ++k) Y[k] = G[k] * sinv;
  Y[0] += sinv; Y[5] += sinv; Y[10] += sinv; Y[15] += sinv;
#pragma unroll
  for (int k = 0; k < 16; ++k) Z[k] = 0.f;
  Z[0] = Z[5] = Z[10] = Z[15] = 1.f;

  for (int it = 0; it < 18; ++it) {
    mm4(Z, Y, T);                       // T = Z*Y
#pragma unroll
    for (int k = 0; k < 16; ++k) W[k] = -0.5f * T[k];
    W[0] += 1.5f; W[5] += 1.5f; W[10] += 1.5f; W[15] += 1.5f;  // W = (3I - ZY)/2
    mm4(Y, W, T);
#pragma unroll
    for (int k = 0; k < 16; ++k) Y[k] = T[k];
    mm4(W, Z, T);
#pragma unroll
    for (int k = 0; k < 16; ++k) Z[k] = T[k];
  }

  float rs = rsqrtf(tr);                // un-normalize: A^{-1/2} = Z / sqrt(tr)
  float S[16];
#pragma unroll
  for (int k = 0; k < 16; ++k) S[k] = Z[k] * rs;
  float* sd = invs + (size_t)node * 16;
#pragma unroll
  for (int k = 0; k < 16; ++k) sd[k] = S[k];

  // Diagonal block: clip(S * G * S, -1, 1)
  float U[16], V[16];
  mm4(S, G, U);
  mm4(U, S, V);
  float* ob = out + (size_t)(4 * node) * N + 4 * node;
#pragma unroll
  for (int i = 0; i < 4; ++i)
#pragma unroll
    for (int j = 0; j < 4; ++j)
      ob[(size_t)i * N + j] = fminf(1.f, fmaxf(-1.f, V[i * 4 + j]));
}

// ---------------------------------------------------------------------------
// K4/K5: per-edge T = clip(S_row * (Ma^T Mb) * S_col); write -T to block
// (row,col), then (in a second launch, preserving reference write order)
// -T^T to block (col,row).
// ---------------------------------------------------------------------------
__global__ void offdiag_kernel(const float* __restrict__ maps,
                               const int* __restrict__ eidx,
                               const float* __restrict__ invs,
                               float* __restrict__ out,
                               int E, int twoE, int N, int transposed) {
  int e = blockIdx.x * blockDim.x + threadIdx.x;
  if (e >= E) return;
  int row = eidx[e];
  int col = eidx[twoE + e];

  float Ma[16], Mb[16], T[16], U[16], V[16];
  const float* pa = maps + (size_t)e * 16;
  const float* pb = maps + (size_t)(E + e) * 16;
#pragma unroll
  for (int k = 0; k < 16; ++k) Ma[k] = pa[k];
#pragma unroll
  for (int k = 0; k < 16; ++k) Mb[k] = pb[k];

  mm4_tn(Ma, Mb, T);                    // T = Ma^T * Mb

  float Sr[16], Sc[16];
  const float* psr = invs + (size_t)row * 16;
  const float* psc = invs + (size_t)col * 16;
#pragma unroll
  for (int k = 0; k < 16; ++k) Sr[k] = psr[k];
#pragma unroll
  for (int k = 0; k < 16; ++k) Sc[k] = psc[k];

  mm4(Sr, T, U);
  mm4(U, Sc, V);
#pragma unroll
  for (int k = 0; k < 16; ++k) V[k] = fminf(1.f, fmaxf(-1.f, V[k]));

  if (!transposed) {
    float* ob = out + (size_t)(4 * row) * N + 4 * col;
#pragma unroll
    for (int i = 0; i < 4; ++i)
#pragma unroll
      for (int j = 0; j < 4; ++j)
        ob[(size_t)i * N + j] = -V[i * 4 + j];
  } else {
    float* ob = out + (size_t)(4 * col) * N + 4 * row;
#pragma unroll
    for (int i = 0; i < 4; ++i)
#pragma unroll
      for (int j = 0; j < 4; ++j)
        ob[(size_t)i * N + j] = -V[j * 4 + i];   // -T^T
  }
}

// ---------------------------------------------------------------------------
// Launch: zero(out) -> zero(ws accumulators) -> WMMA Gram+segment-sum ->
// Newton-Schulz inv-sqrt + diag blocks -> off-diag (row,col) -> (col,row).
// ---------------------------------------------------------------------------
extern "C" void kernel_launch(void* const* d_in, const int* in_sizes, int n_in,
                              void* d_out, int out_size, void* d_ws, size_t ws_size,
                              hipStream_t stream) {
  // setup_inputs order: adj_mat(0), degrees(1), maps(2), edge_index(3)
  const float* maps = (const float*)d_in[2];
  const int*   eidx = (const int*)d_in[3];
  int n    = in_sizes[1];          // 3000 nodes
  int twoE = in_sizes[2] / 16;     // 96000 edges (4x4 maps)
  int E    = twoE / 2;             // 48000
  int N    = 4 * n;                // 12000 (output is N x N)

  float* out  = (float*)d_out;
  float* diag = (float*)d_ws;                  // n*16 floats
  float* invs = diag + (size_t)n * 16;         // n*16 floats

  long long n4 = (long long)out_size / 4;
  zero_out_kernel<<<4096, 256, 0, stream>>>(out, n4);

  int dcount = n * 16;
  zero_ws_kernel<<<(dcount + 255) / 256, 256, 0, stream>>>(diag, dcount);

  int waves = (twoE + 3) / 4;                  // 4 edges per wave32
  int gthreads = waves * 32;
  gram_diag_kernel<<<(gthreads + 255) / 256, 256, 0, stream>>>(maps, eidx, diag, twoE);

  invsqrt_diag_kernel<<<(n + 255) / 256, 256, 0, stream>>>(diag, invs, out, n, N);

  offdiag_kernel<<<(E + 255) / 256, 256, 0, stream>>>(maps, eidx, invs, out, E, twoE, N, 0);
  offdiag_kernel<<<(E + 255) / 256, 256, 0, stream>>>(maps, eidx, invs, out, E, twoE, N, 1);
}